// GaussianSplatRenderer2D_52544629899274
// MI455X (gfx1250) — compile-verified
//
#include <hip/hip_runtime.h>
#include <cstdint>
#include <cstddef>

// ---------------------------------------------------------------------------
// GaussianSplatRenderer2D for MI455X (gfx1250, wave32)
//
// Cost model: ~15.9M samples -> ~63M f32 scatter-adds dominate. The RGBD
// accumulation image is 16 MB -> lives entirely in the 192 MB L2, so the
// kernel is bound by L2 atomic throughput. Atomics are issued without return
// (STOREcnt, fire-and-forget, up to 63 in flight per wave). R,G,B,den are
// interleaved in one 16B group so each sample's 4 adds hit one cacheline.
// exp() count reduced 121 -> 22 per gaussian via separable weights.
// Gaussian parameters are staged through the CDNA5 async-to-LDS path
// (GLOBAL_LOAD_ASYNC_TO_LDS_B32/B64, ASYNCcnt + s_wait_asynccnt).
// ---------------------------------------------------------------------------

#define TPB 256
#define KK 11
#define HALF_K 5
#define EPS_COV 1e-9f
#define EPS_S   1e-6f
#define EPS_DEN 1e-6f

#define AS1 __attribute__((address_space(1)))
#define AS3 __attribute__((address_space(3)))

typedef int v2i __attribute__((ext_vector_type(2)));

#if defined(__has_builtin)
#  if __has_builtin(__builtin_amdgcn_global_load_async_to_lds_b32) && \
      __has_builtin(__builtin_amdgcn_global_load_async_to_lds_b64)
#    define USE_ASYNC_LDS 1
#  endif
#endif

__device__ __forceinline__ void wait_async0() {
#if defined(__has_builtin) && __has_builtin(__builtin_amdgcn_s_wait_asynccnt)
    __builtin_amdgcn_s_wait_asynccnt(0);
#else
    asm volatile("s_wait_asynccnt 0" ::: "memory");
#endif
}

#if defined(USE_ASYNC_LDS)
__device__ __forceinline__ void async_g2l_b64(const float* g, float* l) {
    __builtin_amdgcn_global_load_async_to_lds_b64((AS1 v2i*)g, (AS3 v2i*)l, 0, 0);
}
__device__ __forceinline__ void async_g2l_b32(const float* g, float* l) {
    __builtin_amdgcn_global_load_async_to_lds_b32((AS1 int*)g, (AS3 int*)l, 0, 0);
}
#endif

// Agent-scope relaxed f32 add: should lower to GLOBAL_ATOMIC_ADD_F32
// (no return -> STOREcnt path, non-blocking).
__device__ __forceinline__ void atomAddF(float* p, float v) {
    __hip_atomic_fetch_add(p, v, __ATOMIC_RELAXED, __HIP_MEMORY_SCOPE_AGENT);
}

__global__ __launch_bounds__(TPB)
void splat_accum(const float* __restrict__ mean,   // (B,N,2)
                 const float* __restrict__ cov,    // (B,N,2)
                 const float* __restrict__ rgbp,   // (B,N,3)
                 const float* __restrict__ opap,   // (B,N,1)
                 const int*   __restrict__ Hp,
                 const int*   __restrict__ Wp,
                 float*       __restrict__ ws,     // (B*H*W, 4) RGBD interleaved
                 int BN, int pixels_total)
{
    const int tid = blockIdx.x * TPB + threadIdx.x;

    const int H  = *Hp;
    const int W  = *Wp;
    const int HW = H * W;
    const int B  = pixels_total / HW;
    const int N  = BN / B;

    float ux, vy, cx, cy, cr, cg, cb, op;

#if defined(USE_ASYNC_LDS)
    // Stage the 8 per-gaussian params via the CDNA5 async-to-LDS DMA path.
    __shared__ float stage[TPB * 8];
    float* dstg = &stage[(unsigned)threadIdx.x * 8u];
    if (tid < BN) {
        async_g2l_b64(mean + (size_t)tid * 2, dstg + 0);
        async_g2l_b64(cov  + (size_t)tid * 2, dstg + 2);
        async_g2l_b32(rgbp + (size_t)tid * 3 + 0, dstg + 4);
        async_g2l_b32(rgbp + (size_t)tid * 3 + 1, dstg + 5);
        async_g2l_b32(rgbp + (size_t)tid * 3 + 2, dstg + 6);
        async_g2l_b32(opap + (size_t)tid,         dstg + 7);
        wait_async0();
    }
    if (tid >= BN) return;
    ux = dstg[0]; vy = dstg[1]; cx = dstg[2]; cy = dstg[3];
    cr = dstg[4]; cg = dstg[5]; cb = dstg[6]; op = dstg[7];
#else
    if (tid >= BN) return;
    ux = mean[(size_t)tid * 2 + 0];
    vy = mean[(size_t)tid * 2 + 1];
    cx = cov [(size_t)tid * 2 + 0];
    cy = cov [(size_t)tid * 2 + 1];
    cr = rgbp[(size_t)tid * 3 + 0];
    cg = rgbp[(size_t)tid * 3 + 1];
    cb = rgbp[(size_t)tid * 3 + 2];
    op = opap[(size_t)tid];
#endif

    const int b = tid / N;

    const float sx  = sqrtf(fmaxf(cx, EPS_COV));
    const float sy  = sqrtf(fmaxf(cy, EPS_COV));
    const float isx = 1.0f / fmaxf(sx, EPS_S);
    const float isy = 1.0f / fmaxf(sy, EPS_S);

    // Separable weights; fold bounds-mask into wx, mask+opacity into wy.
    float wx[KK]; int px[KK];
    float wy[KK]; int py[KK];
#pragma unroll
    for (int i = 0; i < KK; ++i) {
        const float off = (float)(i - HALF_K);

        // round-half-even matches jnp.round
        const int pxi = (int)rintf(fmaf(off, sx, ux));
        const bool inx = (pxi >= 0) && (pxi < W);
        px[i] = min(max(pxi, 0), W - 1);
        const float gx = off * isx;
        wx[i] = inx ? __expf(-0.5f * gx * gx) : 0.0f;

        const int pyi = (int)rintf(fmaf(off, sy, vy));
        const bool iny = (pyi >= 0) && (pyi < H);
        py[i] = min(max(pyi, 0), H - 1);
        const float gy = off * isy;
        wy[i] = iny ? (op * __expf(-0.5f * gy * gy)) : 0.0f;
    }

    float* img = ws + (size_t)b * (size_t)HW * 4u;

#pragma unroll 1
    for (int j = 0; j < KK; ++j) {
        const float wyj = wy[j];
        if (wyj == 0.0f) continue;           // whole row out of bounds (or opa==0)
        const int rowbase = py[j] * W;
#pragma unroll
        for (int i = 0; i < KK; ++i) {
            const float a = wyj * wx[i];     // alpha = opa * w * mask
            if (a != 0.0f) {
                float* p4 = img + ((size_t)(rowbase + px[i]) << 2);
                atomAddF(p4 + 0, a * cr);
                atomAddF(p4 + 1, a * cg);
                atomAddF(p4 + 2, a * cb);
                atomAddF(p4 + 3, a);
            }
        }
    }
}

__global__ __launch_bounds__(TPB)
void splat_norm(const float* __restrict__ ws,  // (B*H*W, 4)
                float*       __restrict__ out, // (B, 3, H, W)
                const int*   __restrict__ Hp,
                const int*   __restrict__ Wp,
                int pixels_total)
{
    const int p = blockIdx.x * TPB + threadIdx.x;
    if (p >= pixels_total) return;
    const int HW = (*Hp) * (*Wp);

    const float4 acc = ((const float4*)ws)[p];
    const float inv = 1.0f / fmaxf(acc.w, EPS_DEN);

    const int b   = p / HW;
    const int rem = p - b * HW;
    float* ob = out + (size_t)b * 3u * (size_t)HW + rem;
    ob[0]              = acc.x * inv;
    ob[(size_t)HW]     = acc.y * inv;
    ob[(size_t)HW * 2] = acc.z * inv;
}

extern "C" void kernel_launch(void* const* d_in, const int* in_sizes, int n_in,
                              void* d_out, int out_size, void* d_ws, size_t ws_size,
                              hipStream_t stream) {
    const float* mean = (const float*)d_in[0];
    const float* cov  = (const float*)d_in[1];
    const float* rgb  = (const float*)d_in[2];
    const float* opa  = (const float*)d_in[3];
    const int*   Hp   = (const int*)d_in[4];
    const int*   Wp   = (const int*)d_in[5];

    const int BN           = in_sizes[3];      // opa is (B,N,1) -> B*N elements
    const int pixels_total = out_size / 3;     // out is (B,3,H,W)

    // Zero the RGBD accumulation workspace (16 MB for the reference shapes).
    (void)hipMemsetAsync(d_ws, 0, (size_t)pixels_total * 4u * sizeof(float), stream);

    const int gb_acc  = (BN + TPB - 1) / TPB;
    const int gb_norm = (pixels_total + TPB - 1) / TPB;

    splat_accum<<<gb_acc, TPB, 0, stream>>>(mean, cov, rgb, opa, Hp, Wp,
                                            (float*)d_ws, BN, pixels_total);
    splat_norm<<<gb_norm, TPB, 0, stream>>>((const float*)d_ws, (float*)d_out,
                                            Hp, Wp, pixels_total);
}